// InvertibleFourierGaussianFilter_50955491999784
// MI455X (gfx1250) — compile-verified
//
#include <hip/hip_runtime.h>
#include <cstdint>

// ---------------------------------------------------------------------------
// Fused separable 5x5 Gaussian (FWHM=1.0mm, spacing 1.0) equivalent to the
// reference rfft2/irfft2 pipeline:
//   - X direction: circular convolution (W=1024, power of two -> mask wrap)
//   - Y direction: reflect padding by 2 rows (no wrap reaches after crop)
// Memory-bound op (~806 MB @ 23.3 TB/s => ~35us floor); WMMA is useless here,
// the CDNA5 win is async global->LDS staging + split wait counters.
// ---------------------------------------------------------------------------

#define GLB_AS __attribute__((address_space(1)))
#define LDS_AS __attribute__((address_space(3)))

typedef int v4i __attribute__((vector_size(16)));   // 4 x i32 (b128 payload)

namespace {

constexpr int IMG_H = 768;
constexpr int IMG_W = 1024;
constexpr int TW    = 128;          // tile width  (output)
constexpr int TH    = 32;           // tile height (output)
constexpr int HALO  = 2;
constexpr int RAW_H = TH + 2*HALO;  // 36 rows incl. Y halo
constexpr int RAW_W = TW + 16;      // 144 cols: global [c0-8, c0+135], 16B chunks
constexpr int RAW_CH = RAW_W / 4;   // 36 float4 chunks per row
constexpr int XC_W  = TW;           // x-convolved buffer width

// 1/(2 sigma^2) = 2.35482^2 / 2 = 2.7725886162 (vs ln16 = 2.7725887222)
constexpr double E1  = 0.06250000662746;   // exp(-1 * 2.7725886162)
constexpr double E2  = 1.5258795534e-05;   // exp(-4 * 2.7725886162)
constexpr double SUM = 1.0 + 2.0 * E1 + 2.0 * E2;
constexpr float  W0  = (float)(1.0 / SUM);
constexpr float  W1  = (float)(E1  / SUM);
constexpr float  W2  = (float)(E2  / SUM);

__device__ __forceinline__ void async_copy_b128(const float* gsrc, float* ldst) {
#if __has_builtin(__builtin_amdgcn_global_load_async_to_lds_b128)
    __builtin_amdgcn_global_load_async_to_lds_b128(
        (GLB_AS v4i*)(uintptr_t)gsrc,
        (LDS_AS v4i*)(uint32_t)(uintptr_t)ldst,    // low 32 bits = LDS offset
        0, 0);
#else
    unsigned lds_off = (unsigned)(uintptr_t)ldst;  // aperture low bits = LDS addr
    unsigned long long gaddr = (unsigned long long)(uintptr_t)gsrc;
    asm volatile("global_load_async_to_lds_b128 %0, %1, off"
                 :: "v"(lds_off), "v"(gaddr) : "memory");
#endif
}

__device__ __forceinline__ void wait_async0() {
#if __has_builtin(__builtin_amdgcn_s_wait_asynccnt)
    __builtin_amdgcn_s_wait_asynccnt(0);
#else
    asm volatile("s_wait_asynccnt 0x0" ::: "memory");
#endif
}

__global__ __launch_bounds__(256)
void gauss_fused_kernel(const float* __restrict__ src, float* __restrict__ dst) {
    __shared__ __align__(16) float s_raw[RAW_H * RAW_W]; // 36*144*4 = 20736 B
    __shared__ __align__(16) float s_xc [RAW_H * XC_W];  // 36*128*4 = 18432 B

    const int tid = threadIdx.x;
    const int c0  = blockIdx.x * TW;
    const int r0  = blockIdx.y * TH;
    const size_t img = (size_t)blockIdx.z * (size_t)(IMG_H * IMG_W);
    const float* sb = src + img;
    float*       db = dst + img;

    // ---- Stage 0: async DMA of raw tile (rows with reflect, cols with wrap)
    for (int i = tid; i < RAW_H * RAW_CH; i += 256) {
        const int rr = i / RAW_CH;
        const int q  = i - rr * RAW_CH;
        int gr = r0 - HALO + rr;
        gr = (gr < 0) ? -gr : ((gr >= IMG_H) ? (2 * IMG_H - 2 - gr) : gr);
        const int gc = (c0 - 8 + 4 * q) & (IMG_W - 1);   // circular X, 16B aligned
        async_copy_b128(sb + (size_t)gr * IMG_W + gc, &s_raw[rr * RAW_W + 4 * q]);
    }
    wait_async0();
    __syncthreads();

    // ---- Stage 1: X-pass (circular handled by halo), float4 granularity.
    // Output tile col c maps to raw col c+8; taps need raw cols c+6 .. c+10.
    for (int i = tid; i < RAW_H * (TW / 4); i += 256) {
        const int rr = i >> 5;          // row 0..35 (one row per wave per iter)
        const int q  = i & 31;          // float4 chunk 0..31
        const float4* p = (const float4*)&s_raw[rr * RAW_W + 4 * q + 4];
        const float4 a = p[0], b = p[1], c = p[2];
        const float f[12] = {a.x, a.y, a.z, a.w, b.x, b.y, b.z, b.w,
                             c.x, c.y, c.z, c.w};
        float4 r;
        r.x = W2 * (f[2] + f[6]) + W1 * (f[3] + f[5]) + W0 * f[4];
        r.y = W2 * (f[3] + f[7]) + W1 * (f[4] + f[6]) + W0 * f[5];
        r.z = W2 * (f[4] + f[8]) + W1 * (f[5] + f[7]) + W0 * f[6];
        r.w = W2 * (f[5] + f[9]) + W1 * (f[6] + f[8]) + W0 * f[7];
        *(float4*)&s_xc[rr * XC_W + 4 * q] = r;
    }
    __syncthreads();

    // ---- Stage 2: Y-pass + coalesced float4 stores (wave covers 512B/row).
    for (int i = tid; i < TH * (TW / 4); i += 256) {
        const int rr = i >> 5;          // output row in tile 0..31
        const int q  = i & 31;
        const float4 a0 = *(const float4*)&s_xc[(rr + 0) * XC_W + 4 * q];
        const float4 a1 = *(const float4*)&s_xc[(rr + 1) * XC_W + 4 * q];
        const float4 a2 = *(const float4*)&s_xc[(rr + 2) * XC_W + 4 * q];
        const float4 a3 = *(const float4*)&s_xc[(rr + 3) * XC_W + 4 * q];
        const float4 a4 = *(const float4*)&s_xc[(rr + 4) * XC_W + 4 * q];
        float4 r;
        r.x = W2 * (a0.x + a4.x) + W1 * (a1.x + a3.x) + W0 * a2.x;
        r.y = W2 * (a0.y + a4.y) + W1 * (a1.y + a3.y) + W0 * a2.y;
        r.z = W2 * (a0.z + a4.z) + W1 * (a1.z + a3.z) + W0 * a2.z;
        r.w = W2 * (a0.w + a4.w) + W1 * (a1.w + a3.w) + W0 * a2.w;
        *(float4*)(db + (size_t)(r0 + rr) * IMG_W + (c0 + 4 * q)) = r;
    }
}

} // namespace

extern "C" void kernel_launch(void* const* d_in, const int* in_sizes, int n_in,
                              void* d_out, int out_size, void* d_ws, size_t ws_size,
                              hipStream_t stream) {
    (void)n_in; (void)d_ws; (void)ws_size; (void)out_size;
    const float* x = (const float*)d_in[0];
    float* out = (float*)d_out;
    const int B = in_sizes[0] / (IMG_H * IMG_W);   // 128
    dim3 grid(IMG_W / TW, IMG_H / TH, B);          // (8, 24, 128)
    gauss_fused_kernel<<<grid, 256, 0, stream>>>(x, out);
}